// QNN_68015102099823
// MI455X (gfx1250) — compile-verified
//
#include <hip/hip_runtime.h>
#include <math.h>

// ---------------- problem constants ----------------
#define DDIM   128
#define LDSP   132          // padded LDS row stride (floats): bank = 4*row+col -> conflict-free frags
#define BB     32
#define TT     64
#define INDIM  300
#define DCELL  64
#define NCLS   10
#define BETA_  0.8f
#define THREADS 1024        // 32 waves (wave32)
#define NWAVE  (THREADS/32)
#define NT     2            // output tiles (16x16) per wave: 64 tiles / 32 waves

typedef float v2f __attribute__((ext_vector_type(2)));
typedef float v8f __attribute__((ext_vector_type(8)));

// ---------------- CDNA5 fp32 WMMA: D = A(16x4) x B(4x16) + C(16x16) ----------------
__device__ __forceinline__ v8f wmma4(v2f a, v2f b, v8f c) {
  // 8-arg form: (neg_a, A, neg_b, B, c_mod, C, reuse_a, reuse_b)
  return __builtin_amdgcn_wmma_f32_16x16x4_f32(false, a, false, b, (short)0, c, false, false);
}

// A fragment: 16x4 slice at (m0,k0). lane<16 -> K=k0,k0+1 ; lane>=16 -> K=k0+2,k0+3 ; M = m0+(lane&15)
struct FA {
  const float* p; int ld; int lane;
  __device__ __forceinline__ v2f operator()(int m0, int k0) const {
    const float* q = p + (m0 + (lane & 15)) * ld + k0 + ((lane >> 4) << 1);
    return v2f{q[0], q[1]};
  }
};
// B fragment: 4x16 slice at (k0,n0), row-major B. N = n0+(lane&15)
struct FB {
  const float* p; int ld; int lane;
  __device__ __forceinline__ v2f operator()(int k0, int n0) const {
    int kk = k0 + ((lane >> 4) << 1); int c = n0 + (lane & 15);
    return v2f{p[kk * ld + c], p[(kk + 1) * ld + c]};
  }
};
// B fragment where B = M^T  (i.e. B[k][n] = M[n][k]) -- for the U^dagger side of the sandwich
struct FBT {
  const float* p; int ld; int lane;
  __device__ __forceinline__ v2f operator()(int k0, int n0) const {
    const float* q = p + (n0 + (lane & 15)) * ld + k0 + ((lane >> 4) << 1);
    return v2f{q[0], q[1]};
  }
};

// fragment bundle for one k-step (software pipelining: load next while WMMAs run)
struct Frags {
  v2f a1, a2;
  v2f b1[NT], b2[NT];
};

// OUT(LDS,128x128 @LDSP) = scale * ( A1@B1  +/-  A2@B2 )  [+ OUT if CINIT, scale==1 then]
// Each wave computes NT adjacent tiles in one tile-row => A fragments reused across tiles.
// k-loop is software-pipelined: all fragments for step k are batched, next step prefetched,
// then 2*NT WMMAs issue back-to-back into the XDL pipe.
template<bool NEG2, bool CINIT, class LA1, class LB1, class LA2, class LB2>
__device__ __forceinline__ void gemm2x(float* out, float scale,
                                       LA1 la1, LB1 lb1, LA2 la2, LB2 lb2,
                                       int lane, int wv) {
  const int m0 = ((wv * NT) >> 3) << 4;
  v8f acc[NT];
#pragma unroll
  for (int i = 0; i < NT; ++i) {
    const int n0 = (((wv * NT) + i) & 7) << 4;
    if (CINIT) {
#pragma unroll
      for (int j = 0; j < 8; ++j)
        acc[i][j] = out[(m0 + j + ((lane >> 4) << 3)) * LDSP + n0 + (lane & 15)];
    } else {
      acc[i] = v8f{0.f, 0.f, 0.f, 0.f, 0.f, 0.f, 0.f, 0.f};
    }
  }
  auto loadf = [&](int k0, Frags& f) {
    f.a1 = la1(m0, k0);
    f.a2 = la2(m0, k0);
    if (NEG2) f.a2 = -f.a2;
#pragma unroll
    for (int i = 0; i < NT; ++i) {
      const int n0 = (((wv * NT) + i) & 7) << 4;
      f.b1[i] = lb1(k0, n0);
      f.b2[i] = lb2(k0, n0);
    }
  };
  Frags cur;
  loadf(0, cur);
  for (int k0 = 0; k0 < DDIM; k0 += 4) {
    Frags nxt;
    loadf((k0 + 4 < DDIM) ? (k0 + 4) : 0, nxt);   // prefetch next k-step (wraps harmlessly)
#pragma unroll
    for (int i = 0; i < NT; ++i) {
      acc[i] = wmma4(cur.a1, cur.b1[i], acc[i]);
      acc[i] = wmma4(cur.a2, cur.b2[i], acc[i]);
    }
    cur = nxt;
  }
#pragma unroll
  for (int i = 0; i < NT; ++i) {
    const int n0 = (((wv * NT) + i) & 7) << 4;
#pragma unroll
    for (int j = 0; j < 8; ++j)
      out[(m0 + j + ((lane >> 4) << 3)) * LDSP + n0 + (lane & 15)] = scale * acc[i][j];
  }
}

// ---------------- kernel 1: amplitude embedding -> complex state vectors ----------------
__global__ __launch_bounds__(DDIM) void qrnn_prep(const float* __restrict__ x,
                                                  const float* __restrict__ Wamp,
                                                  const float* __restrict__ bamp,
                                                  const float* __restrict__ phase,
                                                  float* __restrict__ vr,
                                                  float* __restrict__ vi) {
  const int bt = blockIdx.x;     // b*T + t
  const int d  = threadIdx.x;    // 0..127
  const float* xp = x + (size_t)bt * INDIM;
  float a = bamp[d];
  for (int i = 0; i < INDIM; ++i) a += xp[i] * Wamp[i * DDIM + d];  // coalesced over d
  __shared__ float red[DDIM];
  red[d] = a * a;
  __syncthreads();
  for (int s = DDIM / 2; s > 0; s >>= 1) {
    if (d < s) red[d] += red[d + s];
    __syncthreads();
  }
  float nrm = fmaxf(sqrtf(red[0]), 1e-12f);
  float an = a / nrm;
  vr[(size_t)bt * DDIM + d] = an * cosf(phase[d]);
  vi[(size_t)bt * DDIM + d] = an * sinf(phase[d]);
}

// ---------------- kernel 2: full interleaved 2-layer density-matrix scan ----------------
__global__ __launch_bounds__(THREADS) void qrnn_main(
    const float* __restrict__ Uxr, const float* __restrict__ Uxi,
    const float* __restrict__ Uhr, const float* __restrict__ Uhi,
    const float* __restrict__ LamP,
    const float* __restrict__ Kr,  const float* __restrict__ Ki,
    const float* __restrict__ W1,  const float* __restrict__ b1v,
    const float* __restrict__ W2,  const float* __restrict__ b2v,
    const float* __restrict__ vr_all, const float* __restrict__ vi_all,
    float* __restrict__ wsb, float* __restrict__ out) {
  extern __shared__ float smem[];
  float* Nr = smem;                     // pre-activation real (carry working)
  float* Ni = Nr + DDIM * LDSP;
  float* Sr = Ni + DDIM * LDSP;         // temps: lam*(U@H), rho^2, activated rho
  float* Si = Sr + DDIM * LDSP;
  float* wr    = Si + DDIM * LDSP;      // rank-1 state vector (layer 0 input)
  float* wi    = wr + DDIM;
  float* probs = wi + DDIM;             // measurement result
  float* hid   = probs + DDIM;
  float* lgt   = hid + DCELL;
  float* red   = lgt + 16;
  float* scal  = red + THREADS;

  const int tid  = threadIdx.x;
  const int lane = tid & 31;
  const int wv   = tid >> 5;
  const int b    = blockIdx.x;
  const float lam  = LamP[0];
  const float lam1 = 1.f - lam;
  const float bet1 = 1.f - BETA_;

  float* base = wsb + (size_t)b * (6 * DDIM * DDIM);
  float* h1r = base;                   float* h1i = base + 1 * DDIM * DDIM;
  float* h2r = base + 2 * DDIM * DDIM; float* h2i = base + 3 * DDIM * DDIM;
  float* xr  = base + 4 * DDIM * DDIM; float* xi  = base + 5 * DDIM * DDIM;

  // h0 = I/d for both layers (each layer re-initializes, but layers interleave in t)
  for (int idx = tid; idx < DDIM * DDIM; idx += THREADS) {
    float e = ((idx & 127) == (idx >> 7)) ? (1.0f / DDIM) : 0.f;
    h1r[idx] = e; h1i[idx] = 0.f; h2r[idx] = e; h2i[idx] = 0.f;
  }
  __threadfence();
  __syncthreads();

  for (int t = 0; t < TT; ++t) {
    for (int L = 0; L < 2; ++L) {
      float* hr = L ? h2r : h1r;
      float* hi = L ? h2i : h1i;

      // ---- h-sandwich part 1:  S = lam * (U_h @ H)  (complex) ----
      gemm2x<true , false>(Sr, lam, FA{Uhr, DDIM, lane}, FB{hr, DDIM, lane},
                                    FA{Uhi, DDIM, lane}, FB{hi, DDIM, lane}, lane, wv);
      gemm2x<false, false>(Si, lam, FA{Uhr, DDIM, lane}, FB{hi, DDIM, lane},
                                    FA{Uhi, DDIM, lane}, FB{hr, DDIM, lane}, lane, wv);
      __syncthreads();
      // ---- part 2:  N = S @ U_h^dagger  => Nr = Sr@Uhr^T + Si@Uhi^T ; Ni = Si@Uhr^T - Sr@Uhi^T
      gemm2x<false, false>(Nr, 1.f, FA{Sr, LDSP, lane}, FBT{Uhr, DDIM, lane},
                                    FA{Si, LDSP, lane}, FBT{Uhi, DDIM, lane}, lane, wv);
      gemm2x<true , false>(Ni, 1.f, FA{Si, LDSP, lane}, FBT{Uhr, DDIM, lane},
                                    FA{Sr, LDSP, lane}, FBT{Uhi, DDIM, lane}, lane, wv);
      __syncthreads();

      if (L == 0) {
        // rank-1 input:  U_x (v v^dagger) U_x^dagger = w w^dagger,  w = U_x v
        if (tid < DDIM) {
          const float* vrt = vr_all + ((size_t)(b * TT + t)) * DDIM;
          const float* vit = vi_all + ((size_t)(b * TT + t)) * DDIM;
          const float* r0 = Uxr + tid * DDIM;
          const float* r1 = Uxi + tid * DDIM;
          float sr_ = 0.f, si_ = 0.f;
          for (int n = 0; n < DDIM; ++n) {
            float ur = r0[n], ui = r1[n], a_ = vrt[n], c_ = vit[n];
            sr_ += ur * a_ - ui * c_;
            si_ += ur * c_ + ui * a_;
          }
          wr[tid] = sr_; wi[tid] = si_;
        }
        __syncthreads();
        for (int idx = tid; idx < DDIM * DDIM; idx += THREADS) {
          int r = idx >> 7, c = idx & 127;
          Nr[r * LDSP + c] += lam1 * (wr[r] * wr[c] + wi[r] * wi[c]);
          Ni[r * LDSP + c] += lam1 * (wi[r] * wr[c] - wr[r] * wi[c]);
        }
        __syncthreads();
      } else {
        // full x-sandwich on layer-1's activated output (global staging xr/xi)
        gemm2x<true , false>(Sr, lam1, FA{Uxr, DDIM, lane}, FB{xr, DDIM, lane},
                                       FA{Uxi, DDIM, lane}, FB{xi, DDIM, lane}, lane, wv);
        gemm2x<false, false>(Si, lam1, FA{Uxr, DDIM, lane}, FB{xi, DDIM, lane},
                                       FA{Uxi, DDIM, lane}, FB{xr, DDIM, lane}, lane, wv);
        __syncthreads();
        gemm2x<false, true >(Nr, 1.f, FA{Sr, LDSP, lane}, FBT{Uxr, DDIM, lane},
                                      FA{Si, LDSP, lane}, FBT{Uxi, DDIM, lane}, lane, wv);
        gemm2x<true , true >(Ni, 1.f, FA{Si, LDSP, lane}, FBT{Uxr, DDIM, lane},
                                      FA{Sr, LDSP, lane}, FBT{Uxi, DDIM, lane}, lane, wv);
        __syncthreads();
      }

      // new carry is the PRE-activation N (matches the torch loop)
      for (int idx = tid; idx < DDIM * DDIM; idx += THREADS) {
        int r = idx >> 7, c = idx & 127;
        hr[idx] = Nr[r * LDSP + c];
        hi[idx] = Ni[r * LDSP + c];
      }

      // ---- activation: S = N @ N (complex square) ----
      gemm2x<true , false>(Sr, 1.f, FA{Nr, LDSP, lane}, FB{Nr, LDSP, lane},
                                    FA{Ni, LDSP, lane}, FB{Ni, LDSP, lane}, lane, wv);
      gemm2x<false, false>(Si, 1.f, FA{Nr, LDSP, lane}, FB{Ni, LDSP, lane},
                                    FA{Ni, LDSP, lane}, FB{Nr, LDSP, lane}, lane, wv);
      __syncthreads();
      // trace(Sr) with guard
      red[tid] = (tid < DDIM) ? Sr[tid * LDSP + tid] : 0.f;
      __syncthreads();
      for (int s = THREADS / 2; s > 0; s >>= 1) {
        if (tid < s) red[tid] += red[tid + s];
        __syncthreads();
      }
      if (tid == 0) {
        float tr = red[0];
        if (fabsf(tr) < 1e-8f) tr = 1.f;
        scal[0] = BETA_ / tr;
      }
      if (tid < DDIM) probs[tid] = 0.f;
      __syncthreads();
      const float s0 = scal[0];

      if (L == 0) {
        // activated output -> global staging (layer-2 input at this t)
        for (int idx = tid; idx < DDIM * DDIM; idx += THREADS) {
          int r = idx >> 7, c = idx & 127, o = r * LDSP + c;
          xr[idx] = s0 * Sr[o] + bet1 * Nr[o];
          xi[idx] = s0 * Si[o] + bet1 * Ni[o];
        }
      } else {
        // activated output in place (S := activated rho)
        for (int idx = tid; idx < DDIM * DDIM; idx += THREADS) {
          int r = idx >> 7, c = idx & 127, o = r * LDSP + c;
          Sr[o] = s0 * Sr[o] + bet1 * Nr[o];
          Si[o] = s0 * Si[o] + bet1 * Ni[o];
        }
        __syncthreads();
        // ---- fused measurement: probs_k = diag(Re(K^dagger rho K)) ----
        {
          const int m0 = ((wv * NT) >> 3) << 4;
          v8f t1[NT], t2[NT];
#pragma unroll
          for (int i = 0; i < NT; ++i) {
            t1[i] = v8f{0.f, 0.f, 0.f, 0.f, 0.f, 0.f, 0.f, 0.f};
            t2[i] = v8f{0.f, 0.f, 0.f, 0.f, 0.f, 0.f, 0.f, 0.f};
          }
          FA fAr{Sr, LDSP, lane}, fAi{Si, LDSP, lane};
          FB fKr{Kr, DDIM, lane}, fKi{Ki, DDIM, lane};
          // software-pipelined like gemm2x
          struct MFrag { v2f ar, ai, ain; v2f br[NT], bi[NT]; };
          auto mload = [&](int k0, MFrag& f) {
            f.ar = fAr(m0, k0); f.ai = fAi(m0, k0); f.ain = -f.ai;
#pragma unroll
            for (int i = 0; i < NT; ++i) {
              const int n0 = (((wv * NT) + i) & 7) << 4;
              f.br[i] = fKr(k0, n0); f.bi[i] = fKi(k0, n0);
            }
          };
          MFrag cur;
          mload(0, cur);
          for (int k0 = 0; k0 < DDIM; k0 += 4) {
            MFrag nxt;
            mload((k0 + 4 < DDIM) ? (k0 + 4) : 0, nxt);
#pragma unroll
            for (int i = 0; i < NT; ++i) {
              t1[i] = wmma4(cur.ar,  cur.br[i], t1[i]);  // T1 = rho_r@Kr - rho_i@Ki
              t1[i] = wmma4(cur.ain, cur.bi[i], t1[i]);
              t2[i] = wmma4(cur.ar,  cur.bi[i], t2[i]);  // T2 = rho_r@Ki + rho_i@Kr
              t2[i] = wmma4(cur.ai,  cur.br[i], t2[i]);
            }
            cur = nxt;
          }
#pragma unroll
          for (int i = 0; i < NT; ++i) {
            int n0 = (((wv * NT) + i) & 7) << 4;
            int c  = n0 + (lane & 15);
            float p = 0.f;
#pragma unroll
            for (int j = 0; j < 8; ++j) {
              int r = m0 + j + ((lane >> 4) << 3);
              p += Kr[r * DDIM + c] * t1[i][j] + Ki[r * DDIM + c] * t2[i][j];
            }
            atomicAdd(&probs[c], p);   // ds_add_f32 into LDS
          }
        }
        __syncthreads();
        // ---- tiny MLP + log_softmax ----
        if (tid < DCELL) {
          float h_ = b1v[tid];
          for (int d_ = 0; d_ < DDIM; ++d_) h_ += probs[d_] * W1[d_ * DCELL + tid];
          hid[tid] = fmaxf(h_, 0.f);
        }
        __syncthreads();
        if (tid < NCLS) {
          float l_ = b2v[tid];
          for (int j = 0; j < DCELL; ++j) l_ += hid[j] * W2[j * NCLS + tid];
          lgt[tid] = l_;
        }
        __syncthreads();
        if (tid == 0) {
          float mx = lgt[0];
          for (int c = 1; c < NCLS; ++c) mx = fmaxf(mx, lgt[c]);
          float se = 0.f;
          for (int c = 0; c < NCLS; ++c) se += expf(lgt[c] - mx);
          scal[1] = mx + logf(se);
        }
        __syncthreads();
        if (tid < NCLS)
          out[((size_t)(b * TT + t)) * NCLS + tid] = lgt[tid] - scal[1];
      }
      __threadfence();   // make new carry / staging visible to this block's next-iter global reads
      __syncthreads();
    }
  }
}

// ---------------- host-side launch ----------------
extern "C" void kernel_launch(void* const* d_in, const int* in_sizes, int n_in,
                              void* d_out, int out_size, void* d_ws, size_t ws_size,
                              hipStream_t stream) {
  const float* x     = (const float*)d_in[0];
  const float* Wamp  = (const float*)d_in[1];
  const float* bamp  = (const float*)d_in[2];
  const float* phase = (const float*)d_in[3];
  const float* Uxr   = (const float*)d_in[4];
  const float* Uxi   = (const float*)d_in[5];
  const float* Uhr   = (const float*)d_in[6];
  const float* Uhi   = (const float*)d_in[7];
  const float* Lam   = (const float*)d_in[8];
  const float* Kr    = (const float*)d_in[9];
  const float* Ki    = (const float*)d_in[10];
  const float* W1    = (const float*)d_in[11];
  const float* b1v   = (const float*)d_in[12];
  const float* W2    = (const float*)d_in[13];
  const float* b2v   = (const float*)d_in[14];
  float* out = (float*)d_out;
  float* ws  = (float*)d_ws;

  // workspace layout (floats): vr[B*T*D] | vi[B*T*D] | per-batch {h1r,h1i,h2r,h2i,xr,xi} (6*D*D)
  float* vr  = ws;
  float* vi  = ws + (size_t)BB * TT * DDIM;
  float* wsb = ws + (size_t)2 * BB * TT * DDIM;

  qrnn_prep<<<dim3(BB * TT), dim3(DDIM), 0, stream>>>(x, Wamp, bamp, phase, vr, vi);

  constexpr size_t SH_FLOATS = (size_t)4 * DDIM * LDSP   // Nr,Ni,Sr,Si
                             + 2 * DDIM                  // wr,wi
                             + DDIM + DCELL + 16         // probs,hid,lgt
                             + THREADS + 8;              // red,scal
  qrnn_main<<<dim3(BB), dim3(THREADS), SH_FLOATS * sizeof(float), stream>>>(
      Uxr, Uxi, Uhr, Uhi, Lam, Kr, Ki, W1, b1v, W2, b2v, vr, vi, wsb, out);
}